// GCN_3058016715240
// MI455X (gfx1250) — compile-verified
//
#include <hip/hip_runtime.h>
#include <hip/hip_bf16.h>

#define N_NODES 50000
#define N_EDGES 1600000
#define IN_DIM 128
#define HID 64
#define OUT_DIM 16
#define N_GRAPHS 64

typedef float v2f __attribute__((ext_vector_type(2)));
typedef float v8f __attribute__((ext_vector_type(8)));

// ---------------------------------------------------------------------------
// Fill a float4 buffer with a constant (n4 = count of float4 elements).
__global__ void fill4_kernel(float4* __restrict__ p, float v, int n4) {
    int i = blockIdx.x * blockDim.x + threadIdx.x;
    if (i < n4) p[i] = make_float4(v, v, v, v);
}

__global__ void fill_kernel(float* __restrict__ p, float v, int n) {
    int i = blockIdx.x * blockDim.x + threadIdx.x;
    if (i < n) p[i] = v;
}

// ---------------------------------------------------------------------------
// deg[dst] += 1 over edges (deg buffer pre-zeroed).
__global__ void deg_kernel(const int* __restrict__ dst, float* __restrict__ deg, int e) {
    int i = blockIdx.x * blockDim.x + threadIdx.x;
    if (i < e) atomicAdd(&deg[dst[i]], 1.0f);
}

// deg -> dinv = rsqrt(deg + 1) in place.
__global__ void dinv_kernel(float* __restrict__ d, int n) {
    int i = blockIdx.x * blockDim.x + threadIdx.x;
    if (i < n) d[i] = rsqrtf(d[i] + 1.0f);
}

// Per-edge normalization factor, hoisted out of the per-layer scatter.
__global__ void norm_kernel(const int* __restrict__ src, const int* __restrict__ dst,
                            const float* __restrict__ dinv, float* __restrict__ norm, int e) {
    int i = blockIdx.x * blockDim.x + threadIdx.x;
    if (i < e) norm[i] = dinv[src[i]] * dinv[dst[i]];
}

// ---------------------------------------------------------------------------
// H[N,64] = X[N,K] @ W[K,64], fp32 WMMA 16x16x4.
// One wave computes one 16x16 C tile; 4 waves/block cover the 64 output cols.
// A tile (16 x K) staged in LDS with +1 padding to avoid bank conflicts.
template <int K>
__global__ __launch_bounds__(128) void gemm_n64_kernel(const float* __restrict__ X,
                                                       const float* __restrict__ W,
                                                       float* __restrict__ H,
                                                       int nrows) {
    constexpr int LDK = K + 1;
    __shared__ float As[16 * LDK];

    const int row0 = blockIdx.x * 16;
    if (row0 >= nrows) return;
    const int tid = threadIdx.x;

    // Cooperative, coalesced load of the 16 x K A tile.
    for (int i = tid; i < 16 * K; i += 128) {
        int r = i / K;
        int c = i - r * K;
        As[r * LDK + c] = X[(size_t)(row0 + r) * K + c];
    }
    __syncthreads();

    const int wave = tid >> 5;          // 0..3 -> output column tile
    const int lane = tid & 31;
    const int n0   = wave * 16;
    const int l15  = lane & 15;
    const int kh   = (lane >> 4) * 2;   // 0 for lanes 0-15, 2 for lanes 16-31

    v8f c = {};
    const float* wp = W + n0 + l15;     // column of W this lane reads

#pragma unroll 4
    for (int k0 = 0; k0 < K; k0 += 4) {
        v2f a, b;
        // A 16x4 layout: lane<16 holds K=k0..k0+1 of row l15; lane>=16 holds K=k0+2..k0+3
        a[0] = As[l15 * LDK + k0 + kh];
        a[1] = As[l15 * LDK + k0 + kh + 1];
        // B 4x16 layout (row striped across lanes): same K split by lane half
        b[0] = wp[(size_t)(k0 + kh) * 64];
        b[1] = wp[(size_t)(k0 + kh + 1) * 64];
        c = __builtin_amdgcn_wmma_f32_16x16x4_f32(
                /*neg_a=*/false, a, /*neg_b=*/false, b,
                /*c_mod=*/(short)0, c, /*reuse_a=*/false, /*reuse_b=*/false);
    }

    // C/D 16x16 f32 layout: VGPR v holds row v (lanes 0-15) / row v+8 (lanes 16-31)
    float* out = H + (size_t)row0 * 64 + n0;
    const int mbase = (lane >> 4) * 8;
#pragma unroll
    for (int v = 0; v < 8; ++v) {
        out[(size_t)(mbase + v) * 64 + l15] = c[v];
    }
}

// ---------------------------------------------------------------------------
// agg[dst, f] += h[src, f] * norm[e]; 16 threads per edge, float4 per thread.
// total = N_EDGES * 16
__global__ __launch_bounds__(256) void scatter4_kernel(const int* __restrict__ src,
                                                       const int* __restrict__ dst,
                                                       const float* __restrict__ norm,
                                                       const float4* __restrict__ h4,
                                                       float* __restrict__ agg,
                                                       int total) {
    int i = blockIdx.x * blockDim.x + threadIdx.x;
    if (i >= total) return;
    int e = i >> 4;
    int q = i & 15;
    int s = src[e];
    int d = dst[e];
    float nm = norm[e];
    float4 hv = h4[(size_t)s * 16 + q];           // global_load_b128, coalesced per row
    float* ap = agg + (size_t)d * 64 + q * 4;
    atomicAdd(ap + 0, hv.x * nm);
    atomicAdd(ap + 1, hv.y * nm);
    atomicAdd(ap + 2, hv.z * nm);
    atomicAdd(ap + 3, hv.w * nm);
}

// ---------------------------------------------------------------------------
// agg = agg + h * dinv^2 + b  (optionally ReLU), in place; float4 lanes.
// n4 = N_NODES * 16
__global__ __launch_bounds__(256) void finalize4_kernel(float4* __restrict__ agg,
                                                        const float4* __restrict__ h,
                                                        const float* __restrict__ dinv,
                                                        const float4* __restrict__ bias,
                                                        int n4, int relu) {
    int i = blockIdx.x * blockDim.x + threadIdx.x;
    if (i >= n4) return;
    int node = i >> 4;
    int q = i & 15;
    float di = dinv[node];
    float d2 = di * di;
    float4 a = agg[i];
    float4 hv = h[i];
    float4 b = bias[q];
    float4 v;
    v.x = a.x + hv.x * d2 + b.x;
    v.y = a.y + hv.y * d2 + b.y;
    v.z = a.z + hv.z * d2 + b.z;
    v.w = a.w + hv.w * d2 + b.w;
    if (relu) {
        v.x = fmaxf(v.x, 0.0f); v.y = fmaxf(v.y, 0.0f);
        v.z = fmaxf(v.z, 0.0f); v.w = fmaxf(v.w, 0.0f);
    }
    agg[i] = v;
}

// ---------------------------------------------------------------------------
// pooled[batch[n], f] += h[n, f]; cnt[batch[n]] += 1 (once per node); float4 lanes.
__global__ __launch_bounds__(256) void pool4_kernel(const float4* __restrict__ h,
                                                    const int* __restrict__ batch,
                                                    float* __restrict__ pooled,
                                                    float* __restrict__ cnt,
                                                    int n4) {
    int i = blockIdx.x * blockDim.x + threadIdx.x;
    if (i >= n4) return;
    int node = i >> 4;
    int q = i & 15;
    int g = batch[node];
    float4 hv = h[i];
    float* pp = pooled + (size_t)g * 64 + q * 4;
    atomicAdd(pp + 0, hv.x);
    atomicAdd(pp + 1, hv.y);
    atomicAdd(pp + 2, hv.z);
    atomicAdd(pp + 3, hv.w);
    if (q == 0) atomicAdd(&cnt[g], 1.0f);
}

// ---------------------------------------------------------------------------
// out[g, o] = (sum_f pooled[g,f] * Wc[f,o]) / max(cnt[g],1) + bc[o]
__global__ void head_kernel(const float* __restrict__ pooled,
                            const float* __restrict__ cnt,
                            const float* __restrict__ Wc,
                            const float* __restrict__ bc,
                            float* __restrict__ out) {
    int i = blockIdx.x * blockDim.x + threadIdx.x;
    if (i >= N_GRAPHS * OUT_DIM) return;
    int g = i >> 4;
    int o = i & 15;
    float c = fmaxf(cnt[g], 1.0f);
    float acc = 0.0f;
#pragma unroll
    for (int f = 0; f < HID; ++f) acc += pooled[(size_t)g * 64 + f] * Wc[f * OUT_DIM + o];
    out[i] = acc / c + bc[o];
}

// ---------------------------------------------------------------------------
extern "C" void kernel_launch(void* const* d_in, const int* in_sizes, int n_in,
                              void* d_out, int out_size, void* d_ws, size_t ws_size,
                              hipStream_t stream) {
    const float* x   = (const float*)d_in[0];
    const int*   ei  = (const int*)d_in[1];     // [2, E]
    const int*   bat = (const int*)d_in[2];
    const float* W0  = (const float*)d_in[3];
    const float* b0  = (const float*)d_in[4];
    const float* W1  = (const float*)d_in[5];
    const float* b1  = (const float*)d_in[6];
    const float* W2  = (const float*)d_in[7];
    const float* b2  = (const float*)d_in[8];
    const float* Wc  = (const float*)d_in[9];
    const float* bc  = (const float*)d_in[10];
    float* out = (float*)d_out;

    const int* src = ei;
    const int* dst = ei + N_EDGES;

    // Workspace layout (256B-aligned slabs)
    char* base = (char*)d_ws;
    size_t off = 0;
    auto alloc = [&](size_t bytes) {
        void* p = base + off;
        off = (off + bytes + 255) & ~(size_t)255;
        return p;
    };
    float* dinv   = (float*)alloc((size_t)N_NODES * 4);
    float* enorm  = (float*)alloc((size_t)N_EDGES * 4);
    float* buf0   = (float*)alloc((size_t)N_NODES * HID * 4);   // GEMM output h
    float* buf1   = (float*)alloc((size_t)N_NODES * HID * 4);   // aggregation / layer output
    float* pooled = (float*)alloc((size_t)N_GRAPHS * HID * 4);
    float* cnt    = (float*)alloc((size_t)N_GRAPHS * 4);
    (void)ws_size; (void)n_in; (void)in_sizes; (void)out_size;

    const int NF4   = N_NODES * 16;               // float4 elements per feature map
    const int ETOT4 = N_EDGES * 16;               // scatter threads (float4 each)
    const int TB    = 256;
    dim3 blk(TB);
    dim3 g_nodes((N_NODES + TB - 1) / TB);
    dim3 g_nf4((NF4 + TB - 1) / TB);
    dim3 g_edges((N_EDGES + TB - 1) / TB);
    dim3 g_etot4((ETOT4 + TB - 1) / TB);
    dim3 g_gemm(N_NODES / 16);                    // 3125, exact

    // Degrees + per-edge norm (shared by all three layers)
    fill_kernel<<<g_nodes, blk, 0, stream>>>(dinv, 0.0f, N_NODES);
    deg_kernel<<<g_edges, blk, 0, stream>>>(dst, dinv, N_EDGES);
    dinv_kernel<<<g_nodes, blk, 0, stream>>>(dinv, N_NODES);
    norm_kernel<<<g_edges, blk, 0, stream>>>(src, dst, dinv, enorm, N_EDGES);

    // Layer 0: h = x @ W0 ; aggregate ; + self-loop + b0 ; ReLU
    gemm_n64_kernel<IN_DIM><<<g_gemm, dim3(128), 0, stream>>>(x, W0, buf0, N_NODES);
    fill4_kernel<<<g_nf4, blk, 0, stream>>>((float4*)buf1, 0.0f, NF4);
    scatter4_kernel<<<g_etot4, blk, 0, stream>>>(src, dst, enorm, (const float4*)buf0, buf1, ETOT4);
    finalize4_kernel<<<g_nf4, blk, 0, stream>>>((float4*)buf1, (const float4*)buf0, dinv,
                                                (const float4*)b0, NF4, 1);

    // Layer 1
    gemm_n64_kernel<HID><<<g_gemm, dim3(128), 0, stream>>>(buf1, W1, buf0, N_NODES);
    fill4_kernel<<<g_nf4, blk, 0, stream>>>((float4*)buf1, 0.0f, NF4);
    scatter4_kernel<<<g_etot4, blk, 0, stream>>>(src, dst, enorm, (const float4*)buf0, buf1, ETOT4);
    finalize4_kernel<<<g_nf4, blk, 0, stream>>>((float4*)buf1, (const float4*)buf0, dinv,
                                                (const float4*)b1, NF4, 1);

    // Layer 2 (no ReLU)
    gemm_n64_kernel<HID><<<g_gemm, dim3(128), 0, stream>>>(buf1, W2, buf0, N_NODES);
    fill4_kernel<<<g_nf4, blk, 0, stream>>>((float4*)buf1, 0.0f, NF4);
    scatter4_kernel<<<g_etot4, blk, 0, stream>>>(src, dst, enorm, (const float4*)buf0, buf1, ETOT4);
    finalize4_kernel<<<g_nf4, blk, 0, stream>>>((float4*)buf1, (const float4*)buf0, dinv,
                                                (const float4*)b2, NF4, 0);

    // Mean pool + head (pooled and cnt are contiguous -> single fill)
    fill_kernel<<<dim3((N_GRAPHS * HID + N_GRAPHS + TB - 1) / TB), blk, 0, stream>>>(
        pooled, 0.0f, N_GRAPHS * HID + N_GRAPHS);
    pool4_kernel<<<g_nf4, blk, 0, stream>>>((const float4*)buf1, bat, pooled, cnt, NF4);
    head_kernel<<<dim3((N_GRAPHS * OUT_DIM + TB - 1) / TB), blk, 0, stream>>>(
        pooled, cnt, Wc, bc, out);
}